// ExchNetLocalExchange_35055523070250
// MI455X (gfx1250) — compile-verified
//
#include <hip/hip_runtime.h>
#include <hip/hip_bf16.h>
#include <math.h>

#define BATCH   256
#define NCLS    1000
#define ATTN    32
#define CH      512
#define MDIM    (ATTN * CH)      // 16384
#define NTILES  63               // ceil(1000/16)
#define MTILES  (MDIM / 16)      // 1024 (cache kernel handles 2 per wave)

typedef __attribute__((ext_vector_type(2))) float v2f;
typedef __attribute__((ext_vector_type(8))) float v8f;

// ---------------------------------------------------------------------------
// Kernel 1: per-sample argmax class. One block per sample, LDS tree reduction
// with (value desc, index asc) ordering == jnp.argmax first-match semantics.
// ---------------------------------------------------------------------------
__global__ __launch_bounds__(256) void argmax_kernel(
    const float* __restrict__ labels,   // (B, NCLS)
    int* __restrict__ cls)              // (B,)
{
    __shared__ float s_val[256];
    __shared__ int   s_idx[256];
    const int b = blockIdx.x;
    const int t = threadIdx.x;
    const float* row = labels + (size_t)b * NCLS;

    float best = -INFINITY;
    int   bi   = NCLS;
    for (int n = t; n < NCLS; n += 256) {      // ascending n per thread
        float v = row[n];
        if (v > best) { best = v; bi = n; }    // strict > keeps first index
    }
    s_val[t] = best; s_idx[t] = bi;
    __syncthreads();
#pragma unroll
    for (int off = 128; off > 0; off >>= 1) {
        if (t < off) {
            float v2 = s_val[t + off]; int i2 = s_idx[t + off];
            if (v2 > s_val[t] || (v2 == s_val[t] && i2 < s_idx[t])) {
                s_val[t] = v2; s_idx[t] = i2;
            }
        }
        __syncthreads();
    }
    if (t == 0) cls[b] = s_idx[0];
}

// ---------------------------------------------------------------------------
// Kernel 2: deterministic per-class-tile sample lists, padded to multiple of 4.
// Entry packs (class << 16) | sample. Padding = 0xFFFF0000 (class 0xFFFF never
// matches a real class => A contribution exactly 0; sample 0 keeps loads legal).
// ---------------------------------------------------------------------------
__global__ __launch_bounds__(64) void tile_list_kernel(
    const int* __restrict__ cls,
    int* __restrict__ tile_cnt,         // (NTILES,) padded counts
    int* __restrict__ tile_list)        // (NTILES, BATCH)
{
    __shared__ int s_cls[BATCH];
    for (int i = threadIdx.x; i < BATCH; i += 64) s_cls[i] = cls[i];
    __syncthreads();
    const int t = threadIdx.x;
    if (t < NTILES) {
        int cnt = 0;
        const int base = t * BATCH;
        for (int i = 0; i < BATCH; ++i) {           // sequential: bit-stable order
            int cl = s_cls[i];
            if ((cl >> 4) == t) tile_list[base + cnt++] = (cl << 16) | i;
        }
        int padded = (cnt + 3) & ~3;
        for (int j = cnt; j < padded; ++j) tile_list[base + j] = (int)0xFFFF0000;
        tile_cnt[t] = padded;
    }
}

// ---------------------------------------------------------------------------
// Kernel 3: new_count[n] = count[n] + sum_b labels[b,n]
// ---------------------------------------------------------------------------
__global__ __launch_bounds__(256) void count_kernel(
    const float* __restrict__ labels,
    const float* __restrict__ count,
    float* __restrict__ out_count)
{
    int n = blockIdx.x * blockDim.x + threadIdx.x;
    if (n >= NCLS) return;
    float s = count[n];
    for (int b = 0; b < BATCH; ++b) s += labels[(size_t)b * NCLS + n];
    out_count[n] = s;
}

// ---------------------------------------------------------------------------
// Kernel 4: new_cache = cache + einsum('bac,bn->nac') via f32 WMMA with
// K-compaction. One wave32 per (16-class x 32-m) output tile: two 16x16 D
// tiles sharing one A. Inner loop is branch-free (padded lists), EXEC all-1s.
//   C/D 16x16 f32 layout: lane = col (mod 16), VGPR v + 8*(lane>=16) = row.
//   A 16x4 / B 4x16 f32:  lane%16 = M/N index, K = 2*(lane>=16) + vgpr.
// ---------------------------------------------------------------------------
__global__ __launch_bounds__(32) void cache_wmma_kernel(
    const float* __restrict__ features,   // (B, MDIM)
    const float* __restrict__ cache,      // (NCLS, MDIM)
    float* __restrict__ out_cache,        // (NCLS, MDIM)
    const int* __restrict__ tile_cnt,
    const int* __restrict__ tile_list)
{
    const int mtile = blockIdx.x;           // 0..511, covers 32 columns
    const int ntile = blockIdx.y;           // 0..62
    const int lane  = threadIdx.x;
    const int col   = lane & 15;
    const int hi    = lane >> 4;            // 0/1 selects K pair / row half
    const int n0    = ntile * 16;
    const int m0    = mtile * 32;

    // ---- load accumulator = cache tile (uniform fast path for full tiles)
    v8f c0, c1;
    if (n0 + 16 <= NCLS) {
#pragma unroll
        for (int v = 0; v < 8; ++v) {
            const float* p = cache + (size_t)(n0 + v + 8 * hi) * MDIM + m0 + col;
            c0[v] = p[0];
            c1[v] = p[16];
        }
    } else {
#pragma unroll
        for (int v = 0; v < 8; ++v) {
            int  row = n0 + v + 8 * hi;
            bool ok  = row < NCLS;
            const float* p = cache + (size_t)(ok ? row : 0) * MDIM + m0 + col;
            c0[v] = ok ? p[0]  : 0.0f;
            c1[v] = ok ? p[16] : 0.0f;
        }
    }

    const int  kcnt  = tile_cnt[ntile];             // multiple of 4, wave-uniform
    const int* list  = tile_list + ntile * BATCH;
    const int  mycls = n0 + col;                    // class owned by this lane's A slot

    for (int k = 0; k < kcnt; k += 4) {
        // entries for this lane-half's K slots (K = 2*hi + vgpr)
        unsigned e0 = (unsigned)list[k + 2 * hi];
        unsigned e1 = (unsigned)list[k + 2 * hi + 1];

        v2f a, b0, b1;
        a[0] = ((int)(e0 >> 16) == mycls) ? 1.0f : 0.0f;   // exact one-hot label
        a[1] = ((int)(e1 >> 16) == mycls) ? 1.0f : 0.0f;

        const float* f0 = features + (size_t)(e0 & 0xFFFFu) * MDIM + m0 + col;
        const float* f1 = features + (size_t)(e1 & 0xFFFFu) * MDIM + m0 + col;
        b0[0] = f0[0];  b0[1] = f1[0];
        b1[0] = f0[16]; b1[1] = f1[16];

        if (k + 4 < kcnt) {                          // uniform; hide next gather
            unsigned p0 = (unsigned)list[k + 4 + 2 * hi];
            __builtin_prefetch(features + (size_t)(p0 & 0xFFFFu) * MDIM + m0 + col, 0, 1);
        }

        c0 = __builtin_amdgcn_wmma_f32_16x16x4_f32(false, a, false, b0,
                                                   (short)0, c0, false, false);
        c1 = __builtin_amdgcn_wmma_f32_16x16x4_f32(false, a, false, b1,
                                                   (short)0, c1, false, false);
    }

    // ---- store
    if (n0 + 16 <= NCLS) {
#pragma unroll
        for (int v = 0; v < 8; ++v) {
            float* p = out_cache + (size_t)(n0 + v + 8 * hi) * MDIM + m0 + col;
            p[0]  = c0[v];
            p[16] = c1[v];
        }
    } else {
#pragma unroll
        for (int v = 0; v < 8; ++v) {
            int row = n0 + v + 8 * hi;
            if (row < NCLS) {
                float* p = out_cache + (size_t)row * MDIM + m0 + col;
                p[0]  = c0[v];
                p[16] = c1[v];
            }
        }
    }
}

// ---------------------------------------------------------------------------
// Kernel 5: exchange. mask uniform per (b,a); pure b128 gather/copy.
// ---------------------------------------------------------------------------
__global__ __launch_bounds__(128) void exchange_kernel(
    const float* __restrict__ features,   // (B, ATTN, CH)
    const float* __restrict__ anchor,     // (NCLS, ATTN, CH)
    const float* __restrict__ rand_u,     // (B, ATTN)
    const int*   __restrict__ cls,        // (B,)
    float* __restrict__ out_feat)         // (B, ATTN, CH)
{
    const int ba = blockIdx.x;            // b*ATTN + a
    const int b  = ba >> 5;
    const int a  = ba & (ATTN - 1);
    const bool take_anchor = rand_u[ba] > 0.5f;

    const float4* src = take_anchor
        ? (const float4*)(anchor + ((size_t)cls[b] * ATTN + a) * CH)
        : (const float4*)(features + (size_t)ba * CH);
    float4* dst = (float4*)(out_feat + (size_t)ba * CH);
    dst[threadIdx.x] = src[threadIdx.x];  // 128 threads x float4 = 512 floats
}

// ---------------------------------------------------------------------------
extern "C" void kernel_launch(void* const* d_in, const int* in_sizes, int n_in,
                              void* d_out, int out_size, void* d_ws, size_t ws_size,
                              hipStream_t stream)
{
    const float* features = (const float*)d_in[0];   // (256, 32, 512)
    const float* labels   = (const float*)d_in[1];   // (256, 1000)
    const float* anchor   = (const float*)d_in[2];   // (1000, 32, 512)
    const float* cache    = (const float*)d_in[3];   // (1000, 32, 512)
    const float* count    = (const float*)d_in[4];   // (1000,)
    const float* rand_u   = (const float*)d_in[5];   // (256, 32)

    float* out_feat  = (float*)d_out;                         // 256*32*512
    float* out_cache = out_feat + (size_t)BATCH * MDIM;       // 1000*32*512
    float* out_count = out_cache + (size_t)NCLS * MDIM;       // 1000

    // Workspace (ints): cls[256] | tile_cnt[64] | tile_list[63*256]
    int* w_cls  = (int*)d_ws;
    int* w_cnt  = w_cls + BATCH;
    int* w_list = w_cnt + 64;

    argmax_kernel<<<BATCH, 256, 0, stream>>>(labels, w_cls);
    tile_list_kernel<<<1, 64, 0, stream>>>(w_cls, w_cnt, w_list);
    count_kernel<<<(NCLS + 255) / 256, 256, 0, stream>>>(labels, count, out_count);
    cache_wmma_kernel<<<dim3(MTILES / 2, NTILES), 32, 0, stream>>>(
        features, cache, out_cache, w_cnt, w_list);
    exchange_kernel<<<BATCH * ATTN, 128, 0, stream>>>(
        features, anchor, rand_u, w_cls, out_feat);
}